// FeatureExtractor_LSTM_29755533427234
// MI455X (gfx1250) — compile-verified
//
#include <hip/hip_runtime.h>
#include <hip/hip_bf16.h>

// ---------------- problem constants (from reference) ----------------
#define SEQ       8
#define BATCH     16
#define NODE_DIM  8
#define MAX_NODES 512
#define MAX_EDGES 4096
#define HDIM      64      // LSTM hidden == EMB
#define HEADS     2
#define CCH       32      // per-head channels
#define TLAYERS   5
#define NFM_LEN   (NODE_DIM * MAX_NODES)              // 4096
#define FLAT      (NFM_LEN + 2*MAX_EDGES + MAX_NODES + 5)  // 12805
#define NG        (SEQ * BATCH)                       // 128
#define N_TOT     (NG * MAX_NODES)                    // 65536
#define E_G       (NG * MAX_EDGES)                    // 524288
#define E_TOT     (E_G + N_TOT)                       // 589824 (with self loops)
#define NEG_SLOPE 0.2f

// ---------------- WMMA types / helpers ----------------
typedef __attribute__((ext_vector_type(2))) float v2f;
typedef __attribute__((ext_vector_type(8))) float v8f;

// V_WMMA_F32_16X16X4_F32 : D(16x16 f32) = A(16x4 f32) * B(4x16 f32) + C
// 8-arg pattern: (neg_a, A, neg_b, B, c_mod, C, reuse_a, reuse_b)
__device__ __forceinline__ v8f wmma_f32(v2f a, v2f b, v8f c) {
    return __builtin_amdgcn_wmma_f32_16x16x4_f32(false, a, false, b, (short)0, c, false, false);
}

__device__ __forceinline__ float sigmoidf_(float x) { return 1.0f / (1.0f + __expf(-x)); }

// order-preserving float <-> uint mapping for atomicMax-based segment max
__device__ __forceinline__ unsigned fflip(float f) {
    unsigned u = __float_as_uint(f);
    return (u & 0x80000000u) ? ~u : (u | 0x80000000u);
}
__device__ __forceinline__ float funflip(unsigned u) {
    return (u & 0x80000000u) ? __uint_as_float(u & 0x7FFFFFFFu) : __uint_as_float(~u);
}
#define FLIP_NEG_INF 0x007FFFFFu   // fflip(-inf)

// =====================================================================
// LSTM: rows are independent; block owns 16 rows, loops the 8 steps with
// h,c resident in LDS. Gates (16x256) computed with fp32 WMMA.
// grid = 8192/16 = 512 blocks x 256 threads (8 waves; each wave -> 2 gate tiles)
// =====================================================================
__global__ void __launch_bounds__(256)
lstm_kernel(const float* __restrict__ state,
            const float* __restrict__ W_ih,   // [256,8]
            const float* __restrict__ W_hh,   // [256,64]
            const float* __restrict__ b_ih,   // [256]
            const float* __restrict__ b_hh,   // [256]
            float* __restrict__ X)            // [N_TOT,64]
{
    __shared__ float hs[16 * 64];
    __shared__ float cs[16 * 64];
    __shared__ float xt[16 * 8];
    __shared__ float gb[16 * 256];

    const int tid  = threadIdx.x;
    const int wave = tid >> 5;
    const int lane = tid & 31;
    const int ln   = lane & 15;      // A row / B col within tile
    const int kh   = lane >> 4;      // K half-select (0 -> K0,K1 ; 1 -> K2,K3)
    const int r0   = blockIdx.x * 16;

    for (int i = tid; i < 1024; i += 256) { hs[i] = 0.0f; cs[i] = 0.0f; }
    __syncthreads();

    for (int s = 0; s < SEQ; ++s) {
        // stage x_t tile (16 rows x 8 feats)
        for (int i = tid; i < 16 * NODE_DIM; i += 256) {
            int rr = i >> 3, k = i & 7;
            int r = r0 + rr;
            int b = r >> 9, node = r & 511;
            xt[i] = state[(size_t)(s * BATCH + b) * FLAT + node * NODE_DIM + k];
        }
        __syncthreads();

        // gates = x_t @ W_ih^T + h @ W_hh^T + (b_ih + b_hh)
        for (int t = 0; t < 2; ++t) {
            const int g0 = (wave * 2 + t) * 16;
            v8f acc = {0.f, 0.f, 0.f, 0.f, 0.f, 0.f, 0.f, 0.f};

            #pragma unroll
            for (int ks = 0; ks < 2; ++ks) {          // K = 8 (input)
                int k0 = ks * 4 + 2 * kh;
                v2f a = { xt[ln * 8 + k0], xt[ln * 8 + k0 + 1] };
                v2f b = { W_ih[(g0 + ln) * 8 + k0], W_ih[(g0 + ln) * 8 + k0 + 1] };
                acc = wmma_f32(a, b, acc);
            }
            #pragma unroll
            for (int ks = 0; ks < 16; ++ks) {         // K = 64 (hidden)
                int k0 = ks * 4 + 2 * kh;
                v2f a = { hs[ln * 64 + k0], hs[ln * 64 + k0 + 1] };
                v2f b = { W_hh[(g0 + ln) * 64 + k0], W_hh[(g0 + ln) * 64 + k0 + 1] };
                acc = wmma_f32(a, b, acc);
            }
            const float bias = b_ih[g0 + ln] + b_hh[g0 + ln];
            #pragma unroll
            for (int v = 0; v < 8; ++v) {
                int row = (lane < 16) ? v : (v + 8);
                gb[row * 256 + g0 + ln] = acc[v] + bias;
            }
        }
        __syncthreads();

        // gate nonlinearity + state update (torch order i,f,g,o)
        for (int i = tid; i < 1024; i += 256) {
            int row = i >> 6, j = i & 63;
            float gi = gb[row * 256 + j];
            float gf = gb[row * 256 + 64 + j];
            float gg = gb[row * 256 + 128 + j];
            float go = gb[row * 256 + 192 + j];
            float c = sigmoidf_(gf) * cs[i] + sigmoidf_(gi) * tanhf(gg);
            float h = sigmoidf_(go) * tanhf(c);
            cs[i] = c; hs[i] = h;
            X[(size_t)(s * (BATCH * MAX_NODES) + r0 + row) * 64 + j] = h;
        }
        __syncthreads();
    }
}

// =====================================================================
// GAT GEMM: HB = X @ W  (X:[N_TOT,64], W:[64,64])
// grid = N_TOT/16 = 4096 blocks x 128 threads (4 waves; wave -> one 16-col tile)
// =====================================================================
__global__ void __launch_bounds__(128)
gat_gemm_kernel(const float* __restrict__ X, const float* __restrict__ W,
                float* __restrict__ HB)
{
    __shared__ float xs[16 * 64];
    const int tid  = threadIdx.x;
    const int wave = tid >> 5;
    const int lane = tid & 31;
    const int ln   = lane & 15;
    const int kh   = lane >> 4;
    const size_t m0 = (size_t)blockIdx.x * 16;

    __builtin_prefetch(W, 0, 3);   // global_prefetch of the tiny weight matrix

    for (int i = tid; i < 1024; i += 128) xs[i] = X[m0 * 64 + i];
    __syncthreads();

    const int n0 = wave * 16;
    v8f acc = {0.f, 0.f, 0.f, 0.f, 0.f, 0.f, 0.f, 0.f};
    #pragma unroll
    for (int ks = 0; ks < 16; ++ks) {
        int k0 = ks * 4 + 2 * kh;
        v2f a = { xs[ln * 64 + k0], xs[ln * 64 + k0 + 1] };
        v2f b = { W[k0 * 64 + n0 + ln], W[(k0 + 1) * 64 + n0 + ln] };
        acc = wmma_f32(a, b, acc);
    }
    #pragma unroll
    for (int v = 0; v < 8; ++v) {
        int row = (lane < 16) ? v : (v + 8);
        HB[(m0 + row) * 64 + n0 + ln] = acc[v];
    }
}

// =====================================================================
// Edge list construction (graph edges + self loops), batch offsets applied
// =====================================================================
__global__ void build_edges_kernel(const float* __restrict__ state,
                                   int* __restrict__ src, int* __restrict__ dst)
{
    int idx = blockIdx.x * blockDim.x + threadIdx.x;
    if (idx < E_G) {
        int g = idx >> 12, j = idx & 4095;
        const float* p = state + (size_t)g * FLAT + NFM_LEN;
        src[idx] = (int)p[j] + g * MAX_NODES;
        dst[idx] = (int)p[MAX_EDGES + j] + g * MAX_NODES;
    } else if (idx < E_TOT) {
        int n = idx - E_G;
        src[idx] = n;
        dst[idx] = n;
    }
}

// zero aggregation buffer Y (= X, reused), zero Z, init M to -inf (flipped)
__global__ void layer_init_kernel(float* __restrict__ Y, float* __restrict__ Z,
                                  unsigned* __restrict__ M)
{
    int idx = blockIdx.x * blockDim.x + threadIdx.x;
    if (idx < N_TOT * 64) Y[idx] = 0.0f;
    if (idx < N_TOT * HEADS) { Z[idx] = 0.0f; M[idx] = FLIP_NEG_INF; }
}

// logit[e,h] = sum_c a[h,c] * leaky_relu(h[dst]+h[src]); segment max via atomicMax
__global__ void edge_logit_kernel(const int* __restrict__ src, const int* __restrict__ dst,
                                  const float* __restrict__ HB, const float* __restrict__ a,
                                  float* __restrict__ logit, unsigned* __restrict__ M)
{
    int idx = blockIdx.x * blockDim.x + threadIdx.x;
    if (idx >= E_TOT * HEADS) return;
    int e = idx >> 1, h = idx & 1;
    int sN = src[e], dN = dst[e];
    const float* hs_ = HB + (size_t)sN * 64 + h * CCH;
    const float* hd_ = HB + (size_t)dN * 64 + h * CCH;
    const float* av  = a + h * CCH;
    float acc = 0.0f;
    #pragma unroll
    for (int c = 0; c < CCH; ++c) {
        float v = hd_[c] + hs_[c];
        v = (v > 0.0f) ? v : NEG_SLOPE * v;
        acc += av[c] * v;
    }
    logit[idx] = acc;
    atomicMax(&M[dN * 2 + h], fflip(acc));
}

// p = exp(logit - m[dst]); z[dst] += p  (p overwrites logit buffer)
__global__ void edge_p_kernel(const int* __restrict__ dst, float* __restrict__ logit,
                              const unsigned* __restrict__ M, float* __restrict__ Z)
{
    int idx = blockIdx.x * blockDim.x + threadIdx.x;
    if (idx >= E_TOT * HEADS) return;
    int e = idx >> 1, h = idx & 1;
    int dN = dst[e];
    float mv = funflip(M[dN * 2 + h]);
    float pv = __expf(logit[idx] - mv);
    logit[idx] = pv;
    atomicAdd(&Z[dN * 2 + h], pv);
}

// Y[dst,c] += (p / (z[dst]+1e-16)) * HB[src,c]   (thread per (edge, channel))
__global__ void edge_scatter_kernel(const int* __restrict__ src, const int* __restrict__ dst,
                                    const float* __restrict__ HB, const float* __restrict__ P,
                                    const float* __restrict__ Z, float* __restrict__ Y)
{
    int idx = blockIdx.x * blockDim.x + threadIdx.x;
    if (idx >= E_TOT * 64) return;
    int e = idx >> 6, c = idx & 63, h = c >> 5;
    int dN = dst[e], sN = src[e];
    float w = P[e * 2 + h] / (Z[dN * 2 + h] + 1e-16f);
    atomicAdd(&Y[(size_t)dN * 64 + c], w * HB[(size_t)sN * 64 + c]);
}

// x = Y + b  (+ ELU except after last layer); in place on X (== Y)
__global__ void finalize_kernel(float* __restrict__ XY, const float* __restrict__ b,
                                int apply_elu)
{
    int idx = blockIdx.x * blockDim.x + threadIdx.x;
    if (idx >= N_TOT * 64) return;
    float v = XY[idx] + b[idx & 63];
    if (apply_elu) v = (v > 0.0f) ? v : (__expf(v) - 1.0f);
    XY[idx] = v;
}

// feats [N_TOT,67] + scalar N_TOT + valid_entries_idx [NG,2] + num_nodes [N_TOT]
__global__ void assemble_kernel(const float* __restrict__ X, const float* __restrict__ state,
                                float* __restrict__ out)
{
    int n = blockIdx.x * blockDim.x + threadIdx.x;
    if (n >= N_TOT) return;
    float* fr = out + (size_t)n * 67;
    #pragma unroll
    for (int j = 0; j < 64; ++j) fr[j] = X[(size_t)n * 64 + j];
    fr[64] = (float)((n >> 9) & 15);                    // decoupled batch id
    int g = n >> 9, node = n & 511;
    fr[65] = state[(size_t)g * FLAT + NFM_LEN + 2 * MAX_EDGES + node];  // reachable
    float nn = ((n & 8191) < NG) ? (float)MAX_NODES : 0.0f;
    fr[66] = nn;
    const size_t tail = (size_t)N_TOT * 67;
    out[tail + 1 + 2 * NG + n] = nn;                    // num_nodes output
    if (n == 0) out[tail] = (float)N_TOT;               // nodes_in_batch
    if (n < NG) {
        out[tail + 1 + 2 * n]     = (float)(n * MAX_NODES);
        out[tail + 1 + 2 * n + 1] = (float)(n * MAX_NODES + MAX_NODES);
    }
}

// =====================================================================
extern "C" void kernel_launch(void* const* d_in, const int* in_sizes, int n_in,
                              void* d_out, int out_size, void* d_ws, size_t ws_size,
                              hipStream_t stream)
{
    const float* state = (const float*)d_in[0];
    const float* W_ih  = (const float*)d_in[1];
    const float* W_hh  = (const float*)d_in[2];
    const float* b_ih  = (const float*)d_in[3];
    const float* b_hh  = (const float*)d_in[4];
    const float* gat_W = (const float*)d_in[5];   // [T,64,64]
    const float* gat_a = (const float*)d_in[6];   // [T,2,32]
    const float* gat_b = (const float*)d_in[7];   // [T,64]
    float* out = (float*)d_out;

    // workspace layout (floats):  X | HB | SRC | DST | LOGIT | M | Z
    float*    X     = (float*)d_ws;                        // N_TOT*64  (also Y)
    float*    HB    = X + (size_t)N_TOT * 64;              // N_TOT*64
    int*      SRC   = (int*)(HB + (size_t)N_TOT * 64);     // E_TOT
    int*      DST   = SRC + E_TOT;                         // E_TOT
    float*    LOGIT = (float*)(DST + E_TOT);               // E_TOT*2 (then p)
    unsigned* M     = (unsigned*)(LOGIT + (size_t)E_TOT * 2);  // N_TOT*2
    float*    Z     = (float*)(M + (size_t)N_TOT * 2);         // N_TOT*2

    lstm_kernel<<<(BATCH * MAX_NODES) / 16, 256, 0, stream>>>(state, W_ih, W_hh, b_ih, b_hh, X);
    build_edges_kernel<<<(E_TOT + 255) / 256, 256, 0, stream>>>(state, SRC, DST);

    for (int l = 0; l < TLAYERS; ++l) {
        const float* Wl = gat_W + (size_t)l * 64 * 64;
        const float* al = gat_a + (size_t)l * 64;
        const float* bl = gat_b + (size_t)l * 64;
        gat_gemm_kernel<<<N_TOT / 16, 128, 0, stream>>>(X, Wl, HB);
        layer_init_kernel<<<(N_TOT * 64) / 256, 256, 0, stream>>>(X, Z, M);
        edge_logit_kernel<<<(E_TOT * 2) / 256, 256, 0, stream>>>(SRC, DST, HB, al, LOGIT, M);
        edge_p_kernel<<<(E_TOT * 2) / 256, 256, 0, stream>>>(DST, LOGIT, M, Z);
        edge_scatter_kernel<<<(E_TOT * 64) / 256, 256, 0, stream>>>(SRC, DST, HB, LOGIT, Z, X);
        finalize_kernel<<<(N_TOT * 64) / 256, 256, 0, stream>>>(X, bl, (l < TLAYERS - 1) ? 1 : 0);
    }

    assemble_kernel<<<N_TOT / 256, 256, 0, stream>>>(X, state, out);
}